// ImageToGraphTransformer_60601988546908
// MI455X (gfx1250) — compile-verified
//
#include <hip/hip_runtime.h>
#include <math.h>

// Problem constants (reference: B,C,H,W = 32,256,64,64; E=256; heads=8; d=32)
#define BB   32
#define CC   256
#define NN   4096      // H*W
#define EE   256
#define HH   8         // num_heads
#define HD   32        // head_dim
#define INV_SQRT_HD 0.17677669529663687f  // 1/sqrt(32)

typedef __attribute__((ext_vector_type(2))) float f32x2;
typedef __attribute__((ext_vector_type(8))) float f32x8;

// D = A(16x4) * B(4x16) + C(16x16), fp32 exact.
__device__ __forceinline__ f32x8 wmma_f32_k4(f32x2 a, f32x2 b, f32x8 c) {
  return __builtin_amdgcn_wmma_f32_16x16x4_f32(false, a, false, b, (short)0, c,
                                               false, false);
}

// ---------------------------------------------------------------------------
// 1) x_mean[b,c] = mean_n x[b,c,n]   (x row of 4096 floats is contiguous)
// ---------------------------------------------------------------------------
__global__ void mean_kernel(const float* __restrict__ x, float* __restrict__ xm) {
  const int row = blockIdx.x;              // b*C + c
  const int t = threadIdx.x;               // 256 threads
  const float4* p = (const float4*)(x + (size_t)row * NN);
  float s = 0.0f;
#pragma unroll
  for (int i = 0; i < 4; ++i) {
    float4 v = p[i * 256 + t];
    s += v.x + v.y + v.z + v.w;
  }
  __shared__ float red[256];
  red[t] = s;
  __syncthreads();
  for (int st = 128; st > 0; st >>= 1) {
    if (t < st) red[t] += red[t + st];
    __syncthreads();
  }
  if (t == 0) xm[row] = red[0] * (1.0f / (float)NN);
}

// ---------------------------------------------------------------------------
// 2) qs[b,e] = (x_mean[b,:]·Wq[e,:] + bq[e]) / sqrt(head_dim)
//    block 0 also fills the 'batch' output and penalty=0.
// ---------------------------------------------------------------------------
__global__ void qfold_kernel(const float* __restrict__ xm,
                             const float* __restrict__ Wq,
                             const float* __restrict__ bq,
                             float* __restrict__ qs,
                             float* __restrict__ out_batch,
                             float* __restrict__ out_pen) {
  const int b = blockIdx.x, t = threadIdx.x;
  __shared__ float xml[CC];
  xml[t] = xm[b * CC + t];
  __syncthreads();
  float acc = bq[t];
  const float* wr = Wq + (size_t)t * CC;
  for (int c = 0; c < CC; ++c) acc += xml[c] * wr[c];
  qs[b * EE + t] = acc * INV_SQRT_HD;
  if (b == 0) {
    out_batch[t] = (float)(t >> 3);      // repeat(arange(B), heads)
    if (t == 0) out_pen[0] = 0.0f;
  }
}

// ---------------------------------------------------------------------------
// 3) Fold q into Wk:  wqk[b,h,c] = sum_{j<32} qs[b,h*32+j]*Wk[h*32+j,c]
//                     beta[b,h]  = sum_{j<32} qs[b,h*32+j]*bk[h*32+j]
// ---------------------------------------------------------------------------
__global__ void wqk_kernel(const float* __restrict__ qs,
                           const float* __restrict__ Wk,
                           const float* __restrict__ bk,
                           float* __restrict__ wqk,
                           float* __restrict__ beta) {
  const int bh = blockIdx.x, t = threadIdx.x;   // t = c
  const int b = bh >> 3, h = bh & 7;
  __shared__ float q32[HD];
  if (t < HD) q32[t] = qs[b * EE + h * HD + t];
  __syncthreads();
  float acc = 0.0f;
#pragma unroll 4
  for (int j = 0; j < HD; ++j) acc += q32[j] * Wk[(size_t)(h * HD + j) * CC + t];
  wqk[(size_t)bh * CC + t] = acc;
  if (t == 0) {
    float bb = 0.0f;
    for (int j = 0; j < HD; ++j) bb += q32[j] * bk[h * HD + j];
    beta[bh] = bb;
  }
}

// ---------------------------------------------------------------------------
// 4) logits[b,h,n] = sum_c wqk[b,h,c]*x[b,c,n] + beta[b,h]   (WMMA f32)
//    A = wqk (M=16 padded heads, K=c), B = x (K=c, N=n), per-wave 16-col tile.
//    grid = (NN/128, B), block = 256 (8 waves)
// ---------------------------------------------------------------------------
__global__ void logits_kernel(const float* __restrict__ x,
                              const float* __restrict__ wqk,
                              const float* __restrict__ beta,
                              float* __restrict__ logits) {
  __shared__ float As[16 * 257];               // 257 stride -> conflict-free
  const int b = blockIdx.y;
  const int t = threadIdx.x;
  const float* wq = wqk + (size_t)b * HH * CC;
#pragma unroll
  for (int m = 0; m < 16; ++m)
    As[m * 257 + t] = (m < HH) ? wq[m * CC + t] : 0.0f;
  __syncthreads();

  const int wave = t >> 5;
  const int lane = t & 31;
  const int l16  = lane & 15;
  const int koff = (lane >> 4) << 1;           // 0 or 2
  const int nb = blockIdx.x * 128 + wave * 16; // 16 spatial columns per wave
  const float* xb = x + (size_t)b * CC * NN;

  f32x8 acc = {};
  for (int k = 0; k < CC; k += 4) {
    f32x2 a, bf;
    a.x = As[l16 * 257 + k + koff];
    a.y = As[l16 * 257 + k + koff + 1];
    bf.x = xb[(size_t)(k + koff) * NN + nb + l16];
    bf.y = xb[(size_t)(k + koff + 1) * NN + nb + l16];
    acc = wmma_f32_k4(a, bf, acc);
  }
  // D: VGPR r, lanes 0-15 hold M=r (heads 0..7); lanes 16-31 hold padding rows.
  if (lane < 16) {
#pragma unroll
    for (int r = 0; r < HH; ++r)
      logits[((size_t)b * HH + r) * NN + nb + l16] = acc[r] + beta[b * HH + r];
  }
}

// ---------------------------------------------------------------------------
// 5) Softmax over N per (b,h); writes probs back into ws (layout [b,h,n]),
//    writes attention output ([b,n,h]) and pos output.
// ---------------------------------------------------------------------------
__global__ void softmax_kernel(float* __restrict__ logits,
                               float* __restrict__ attn_out,
                               float* __restrict__ pos_out) {
  const int bh = blockIdx.x;
  const int b = bh >> 3, h = bh & 7;
  const int t = threadIdx.x;
  float* row = logits + (size_t)bh * NN;
  __shared__ float red[256];

  float vals[16];
  float mx = -INFINITY;
#pragma unroll
  for (int i = 0; i < 16; ++i) {
    vals[i] = row[i * 256 + t];
    mx = fmaxf(mx, vals[i]);
  }
  red[t] = mx;
  __syncthreads();
  for (int s = 128; s > 0; s >>= 1) {
    if (t < s) red[t] = fmaxf(red[t], red[t + s]);
    __syncthreads();
  }
  mx = red[0];
  __syncthreads();

  float sum = 0.0f;
#pragma unroll
  for (int i = 0; i < 16; ++i) {
    vals[i] = __expf(vals[i] - mx);
    sum += vals[i];
  }
  red[t] = sum;
  __syncthreads();
  for (int s = 128; s > 0; s >>= 1) {
    if (t < s) red[t] += red[t + s];
    __syncthreads();
  }
  const float inv = 1.0f / red[0];
  __syncthreads();

  float px = 0.0f, py = 0.0f;
#pragma unroll
  for (int i = 0; i < 16; ++i) {
    const int n = i * 256 + t;
    const float p = vals[i] * inv;
    row[n] = p;                                      // probs, [b,h,n] for pass 2
    attn_out[((size_t)b * NN + n) * HH + h] = p;     // output layout [B,N,h]
    px += p * (float)(n >> 6);                       // coords: (n/64, n%64)
    py += p * (float)(n & 63);
  }
  red[t] = px;
  __syncthreads();
  for (int s = 128; s > 0; s >>= 1) {
    if (t < s) red[t] += red[t + s];
    __syncthreads();
  }
  if (t == 0) pos_out[bh * 2 + 0] = red[0];
  __syncthreads();
  red[t] = py;
  __syncthreads();
  for (int s = 128; s > 0; s >>= 1) {
    if (t < s) red[t] += red[t + s];
    __syncthreads();
  }
  if (t == 0) pos_out[bh * 2 + 1] = red[0];
}

// ---------------------------------------------------------------------------
// 6) xa[b,h,c] = sum_n attn[b,h,n]*x[b,c,n]    (WMMA f32, K=n)
//    A = x tile (M=16 c's, K=n), B = probs (K=n, N=16 padded heads).
//    grid = (C/16, B), block = 256 (8 waves split the n-range), LDS reduce.
// ---------------------------------------------------------------------------
__global__ void xa_kernel(const float* __restrict__ x,
                          const float* __restrict__ probs,
                          float* __restrict__ xa) {
  const int b = blockIdx.y;
  const int cbase = blockIdx.x * 16;
  const int t = threadIdx.x;
  const int wave = t >> 5, lane = t & 31;
  const int l16 = lane & 15;
  const int koff = (lane >> 4) << 1;
  const float* xb = x + (size_t)b * CC * NN;
  const float* ap = probs + (size_t)b * HH * NN;
  const float hv = (l16 < HH) ? 1.0f : 0.0f;       // zero pad heads 8..15
  const int hrow = l16 & 7;                        // in-bounds row for loads

  f32x8 acc = {};
  const int n0 = wave * (NN / 8);
  for (int k = n0; k < n0 + NN / 8; k += 4) {
    f32x2 a = *(const f32x2*)(xb + (size_t)(cbase + l16) * NN + k + koff);
    f32x2 pv = *(const f32x2*)(ap + (size_t)hrow * NN + k + koff);
    f32x2 bf;
    bf.x = pv.x * hv;
    bf.y = pv.y * hv;
    acc = wmma_f32_k4(a, bf, acc);
  }

  __shared__ float red[8 * 256];
#pragma unroll
  for (int r = 0; r < 8; ++r) red[wave * 256 + r * 32 + lane] = acc[r];
  __syncthreads();
  float s = 0.0f;
#pragma unroll
  for (int w = 0; w < 8; ++w) s += red[w * 256 + t];
  // element t -> (r = t/32, lane2 = t%32): M(c_local)=r+8*(lane2/16), N(head)=lane2%16
  const int r2 = t >> 5, lane2 = t & 31;
  const int cl = r2 + ((lane2 >> 4) << 3);
  const int hh = lane2 & 15;
  if (hh < HH) xa[((size_t)b * HH + hh) * CC + cbase + cl] = s;
}

// ---------------------------------------------------------------------------
// 7) values[m,e] = sum_c xa[m,c]*Wv[e,c] + bv[e],  m = b*8+h  (WMMA f32)
//    256x256x256 GEMM: 256 tiles of 16x16, 8 waves/block -> 32 blocks.
// ---------------------------------------------------------------------------
__global__ void values_kernel(const float* __restrict__ xa,
                              const float* __restrict__ Wv,
                              const float* __restrict__ bv,
                              float* __restrict__ out_values) {
  const int t = threadIdx.x;
  const int wave = t >> 5, lane = t & 31;
  const int l16 = lane & 15;
  const int koff = (lane >> 4) << 1;
  const int tile = blockIdx.x * 8 + wave;
  const int mb = (tile >> 4) * 16;   // rows (b*h)
  const int nb = (tile & 15) * 16;   // cols (e)

  f32x8 acc = {};
  for (int k = 0; k < CC; k += 4) {
    f32x2 a = *(const f32x2*)(xa + (size_t)(mb + l16) * CC + k + koff);
    f32x2 bf = *(const f32x2*)(Wv + (size_t)(nb + l16) * CC + k + koff);
    acc = wmma_f32_k4(a, bf, acc);
  }
  const float bias = bv[nb + l16];
#pragma unroll
  for (int r = 0; r < 8; ++r) {
    const int m = r + ((lane >> 4) << 3);
    out_values[(size_t)(mb + m) * EE + nb + l16] = acc[r] + bias;
  }
}

// ---------------------------------------------------------------------------
extern "C" void kernel_launch(void* const* d_in, const int* in_sizes, int n_in,
                              void* d_out, int out_size, void* d_ws, size_t ws_size,
                              hipStream_t stream) {
  const float* x  = (const float*)d_in[0];
  const float* Wq = (const float*)d_in[1];
  const float* bq = (const float*)d_in[2];
  const float* Wk = (const float*)d_in[3];
  const float* bk = (const float*)d_in[4];
  const float* Wv = (const float*)d_in[5];
  const float* bv = (const float*)d_in[6];
  (void)in_sizes; (void)n_in; (void)out_size; (void)ws_size;

  float* out = (float*)d_out;
  float* out_values = out;                 // [256,256]
  float* out_pos    = out + 65536;         // [256,2]
  float* out_batch  = out + 66048;         // [256]
  float* out_attn   = out + 66304;         // [32,4096,8]
  float* out_pen    = out + 1114880;       // scalar

  float* ws   = (float*)d_ws;
  float* xm   = ws;                        // 8192
  float* qs   = ws + 8192;                 // 8192
  float* wqk  = ws + 16384;                // 65536
  float* beta = ws + 81920;                // 256
  float* xa   = ws + 82176;                // 65536
  float* lgts = ws + 147712;               // 1048576 (logits -> probs)

  mean_kernel<<<BB * CC, 256, 0, stream>>>(x, xm);
  qfold_kernel<<<BB, 256, 0, stream>>>(xm, Wq, bq, qs, out_batch, out_pen);
  wqk_kernel<<<BB * HH, 256, 0, stream>>>(qs, Wk, bk, wqk, beta);
  logits_kernel<<<dim3(NN / 128, BB), 256, 0, stream>>>(x, wqk, beta, lgts);
  softmax_kernel<<<BB * HH, 256, 0, stream>>>(lgts, out_attn, out_pos);
  xa_kernel<<<dim3(CC / 16, BB), 256, 0, stream>>>(x, lgts, xa);
  values_kernel<<<32, 256, 0, stream>>>(xa, Wv, bv, out_values);
}